// JacobiToCartesian_58995670777993
// MI455X (gfx1250) — compile-verified
//
#include <hip/hip_runtime.h>
#include <hip/hip_bf16.h>

// Problem constants (from reference): B=128, T=4096, N=16, D=3
#define B_ 128
#define T_ 4096
#define N_ 16
#define D_ 3

typedef float v2f __attribute__((ext_vector_type(2)));
typedef float v8f __attribute__((ext_vector_type(8)));

// ---------------------------------------------------------------------------
// Kernel 1: build Binv = inv(A) analytically from the Jacobi-coordinate
// structure of A.  Closed form (verified A*Binv = I):
//   Binv[i][0] = 1
//   Binv[i][i] = M_{i-1}/M_i          (i >= 1)
//   Binv[i][j] = -m_j / M_j           (j > i, and all of row 0 for j>=1)
//   Binv[i][j] = 0                    (1 <= j < i)
// where M_k = cumsum(m)[k].
// Grid: B blocks x 256 threads; thread t computes entry (i=t/16, j=t%16).
// ---------------------------------------------------------------------------
__global__ __launch_bounds__(256) void build_binv_kernel(
    const float* __restrict__ m, float* __restrict__ binv) {
  const int b = blockIdx.x;
  const int t = threadIdx.x;
  const int i = t >> 4;
  const int j = t & 15;
  const float* mb = m + b * N_;

  float run = 0.f, Mj = 0.f, Mi = 0.f, Mim1 = 0.f;
#pragma unroll
  for (int k = 0; k < N_; ++k) {
    float prev = run;
    run += mb[k];
    if (k == j) Mj = run;
    if (k == i) { Mi = run; Mim1 = prev; }
  }

  float val;
  if (j == 0)               val = 1.0f;
  else if (i == 0 || j > i) val = -mb[j] / Mj;
  else if (j == i)          val = Mim1 / Mi;
  else                      val = 0.0f;

  binv[b * 256 + i * 16 + j] = val;
}

// ---------------------------------------------------------------------------
// Kernel 2: streaming apply.  out[b,t,i,d] = sum_j Binv[b,i,j] * x[b,t,j,d]
// Each wave handles one chunk = 16 timesteps of one batch = 768 contiguous
// floats per source.  WMMA tiles: A = Binv (16x16, chained 4x K=4 f32 WMMA),
// B = X[j, t_local] for fixed d (3 column groups).
//
// Register layouts used (wave32):
//  A 16x4 f32:  lane L (M = L&15), VGPR v: K = 4c + v + 2*(L>>4)  [chain c]
//  B 4x16 f32:  lane L (N = L&15), VGPR v: K = 4c + v + 2*(L>>4)
//  C/D 16x16:   VGPR r: M = r + 8*(L>>4), N = L&15
// ---------------------------------------------------------------------------
__global__ __launch_bounds__(128) void apply_binv_kernel(
    const float* __restrict__ binv,
    const float* __restrict__ qj,
    const float* __restrict__ vj,
    float* __restrict__ outq,
    float* __restrict__ outv) {
  __shared__ float lds[4 * 1536];  // per wave: 768 in + 768 out

  const int wave = threadIdx.x >> 5;
  const int lane = threadIdx.x & 31;
  const int col  = lane & 15;      // B/D column (t_local), A/D row helper
  const int h    = lane >> 4;      // lane half

  const int gw = blockIdx.x * 4 + wave;  // global chunk id, 0..32767
  const int b  = gw >> 8;                // 256 chunks per batch
  const int t0 = (gw & 255) << 4;        // 16 timesteps per chunk

  float* inBuf  = lds + wave * 1536;
  float* outBuf = inBuf + 768;

  // Per-batch Binv into A-matrix register layout (8 scalar loads, reused for
  // the whole chunk: 24 WMMA chains share these regs).
  v2f a[4];
  const float* bb = binv + b * 256 + col * 16;
#pragma unroll
  for (int c = 0; c < 4; ++c) {
    a[c].x = bb[4 * c + 0 + 2 * h];
    a[c].y = bb[4 * c + 1 + 2 * h];
  }

  const size_t chunkBase = ((size_t)b * T_ + t0) * (N_ * D_);  // floats
  const float* srcs[2] = { qj + chunkBase, vj + chunkBase };
  float*       dsts[2] = { outq + chunkBase, outv + chunkBase };

  // Hint the v-chunk into cache while we chew on q.
  __builtin_prefetch(vj + chunkBase, 0, 1);

#pragma unroll
  for (int s = 0; s < 2; ++s) {
    const float4* gsrc = (const float4*)srcs[s];
    float4*       gdst = (float4*)dsts[s];

    // Stage 768 floats (192 float4) into LDS, fully coalesced b128 loads.
#pragma unroll
    for (int i = 0; i < 6; ++i) {
      ((float4*)inBuf)[i * 32 + lane] = gsrc[i * 32 + lane];
    }
    __syncthreads();

    // One WMMA chain per Cartesian dim d = g (columns are the 16 t_locals).
#pragma unroll
    for (int g = 0; g < 3; ++g) {
      v8f acc = {};
#pragma unroll
      for (int c = 0; c < 4; ++c) {
        const int j0 = 4 * c + 2 * h;
        v2f bm;
        bm.x = inBuf[col * 48 + (j0 + 0) * 3 + g];
        bm.y = inBuf[col * 48 + (j0 + 1) * 3 + g];
        // D = A x B + C ; chain K=4 slices to K=16.
        acc = __builtin_amdgcn_wmma_f32_16x16x4_f32(
            false, a[c], false, bm, (short)0, acc, false, false);
      }
      // Scatter D: VGPR r holds (M = r + 8h, N = col).
#pragma unroll
      for (int r = 0; r < 8; ++r) {
        outBuf[col * 48 + (r + 8 * h) * 3 + g] = acc[r];
      }
    }
    __syncthreads();

    // Coalesced b128 store of the transformed chunk.
#pragma unroll
    for (int i = 0; i < 6; ++i) {
      gdst[i * 32 + lane] = ((float4*)outBuf)[i * 32 + lane];
    }
    __syncthreads();
  }
}

// ---------------------------------------------------------------------------
extern "C" void kernel_launch(void* const* d_in, const int* in_sizes, int n_in,
                              void* d_out, int out_size, void* d_ws, size_t ws_size,
                              hipStream_t stream) {
  const float* m  = (const float*)d_in[0];   // (B, N)
  const float* qj = (const float*)d_in[1];   // (B, T, N, D)
  const float* vj = (const float*)d_in[2];   // (B, T, N, D)

  float* outq = (float*)d_out;
  float* outv = outq + (size_t)B_ * T_ * N_ * D_;

  float* binv = (float*)d_ws;  // B * 16 * 16 floats = 128 KB scratch

  build_binv_kernel<<<B_, 256, 0, stream>>>(m, binv);

  const int chunks = B_ * (T_ / 16);        // 32768 chunks
  apply_binv_kernel<<<chunks / 4, 128, 0, stream>>>(binv, qj, vj, outq, outv);
}